// Mamba_6270652252312
// MI455X (gfx1250) — compile-verified
//
#include <hip/hip_runtime.h>

// ---------------------------------------------------------------------------
// Problem constants (from the reference)
// ---------------------------------------------------------------------------
#define B_    128
#define L_    63
#define BL_   8064          // B_*L_
#define DM_   100           // D_MODEL
#define NS_   300           // N_STATE

#define ACT_NONE     0
#define ACT_LRELU    1
#define ACT_SOFTPLUS 2

typedef float v2f __attribute__((ext_vector_type(2)));
typedef float v8f __attribute__((ext_vector_type(8)));

__device__ __forceinline__ float wave_sum32(float v) {
#pragma unroll
  for (int off = 16; off > 0; off >>= 1) v += __shfl_xor(v, off, 32);
  return v;
}

__device__ __forceinline__ int imin(int a, int b) { return a < b ? a : b; }

// ---------------------------------------------------------------------------
// fp32 WMMA GEMM: C[M,N] = act(A[M,K] @ W[K,N] + bias[N])
// Each wave computes a 16x64 strip (4 accumulators) with V_WMMA_F32_16X16X4_F32;
// block = 4 waves -> 64x64 macro tile.
// All hot-loop indices are *unsigned* 32-bit (base SGPR pointer + zext(u32)
// voffset == the hardware GVS addressing mode), per-lane offsets are
// loop-invariant, and one uniform kN advances per k-step -> one 32-bit VALU
// add per load, no 64-bit address math.  Out-of-range rows/cols use clamped
// (valid) addresses and are discarded by store guards; only the K-tail
// (K%4!=0) zeroes A values with selects.  Branch-free body.
// f32 fragments: A(16x4): lane m=(lane&15), VGPR0 K=k+(lane>=16?2:0), VGPR1 K+1.
// B(4x16) mirrors with lane = column.  C/D: VGPR v -> row v+(lane>=16?8:0).
// ---------------------------------------------------------------------------
__global__ void wmma_gemm_bias_act(const float* __restrict__ A,
                                   const float* __restrict__ W,
                                   const float* __restrict__ bias,
                                   float* __restrict__ C,
                                   int M, int N, int K, int act)
{
  const int lane  = threadIdx.x & 31;
  const int wave  = threadIdx.x >> 5;
  const int col0  = blockIdx.x * 64;
  const int row0  = blockIdx.y * 64 + wave * 16;
  if (row0 >= M) return;                       // wave-uniform

  const int laneN = lane & 15;
  const int koff  = (lane >> 4) << 1;          // lanes 16..31: K+2/K+3
  const int mA    = imin(row0 + laneN, M - 1); // clamped row
  const unsigned aBase = (unsigned)(mA * K + koff);  // per-lane A index base

  unsigned wB0[4], wB1[4];
  int nb[4];
#pragma unroll
  for (int t = 0; t < 4; ++t) {
    nb[t]  = imin(col0 + 16 * t + laneN, N - 1);     // clamped col
    wB0[t] = (unsigned)(koff * N + nb[t]);           // fixed per-lane offsets
    wB1[t] = wB0[t] + (unsigned)N;
  }

  v8f acc[4] = {{}, {}, {}, {}};
  const int K4 = K & ~3;
  unsigned kN = 0;
  for (int k = 0; k < K4; k += 4) {
    v2f a;
    a.x = A[aBase + (unsigned)k];
    a.y = A[aBase + (unsigned)k + 1u];
    v2f bf[4];
#pragma unroll
    for (int t = 0; t < 4; ++t) {
      bf[t].x = W[kN + wB0[t]];
      bf[t].y = W[kN + wB1[t]];
    }
#pragma unroll
    for (int t = 0; t < 4; ++t)
      acc[t] = __builtin_amdgcn_wmma_f32_16x16x4_f32(false, a, false, bf[t],
                                                     (short)0, acc[t],
                                                     false, false);
    kN += 4u * (unsigned)N;
  }
  if (K4 < K) {                                // tail: zero invalid K slots in A
    const int ka   = K4 + koff;
    const int kac0 = imin(ka, K - 1);
    const int kac1 = imin(ka + 1, K - 1);
    v2f a;
    a.x = A[(unsigned)(mA * K + kac0)];  a.x = (ka     < K) ? a.x : 0.0f;
    a.y = A[(unsigned)(mA * K + kac1)];  a.y = (ka + 1 < K) ? a.y : 0.0f;
    const unsigned r0 = (unsigned)(kac0 * N);
    const unsigned r1 = (unsigned)(kac1 * N);
    v2f bf[4];
#pragma unroll
    for (int t = 0; t < 4; ++t) {
      bf[t].x = W[r0 + (unsigned)nb[t]];
      bf[t].y = W[r1 + (unsigned)nb[t]];
    }
#pragma unroll
    for (int t = 0; t < 4; ++t)
      acc[t] = __builtin_amdgcn_wmma_f32_16x16x4_f32(false, a, false, bf[t],
                                                     (short)0, acc[t],
                                                     false, false);
  }

  const int  mbase = row0 + ((lane >> 4) << 3);
  const bool mOK   = (row0 + 15 < M);          // true for all calls here (M%64==0)
#pragma unroll
  for (int t = 0; t < 4; ++t) {
    const int n = col0 + 16 * t + laneN;       // unclamped for the guard
    if (n < N) {
      const float bv = bias ? bias[n] : 0.0f;
#pragma unroll
      for (int v = 0; v < 8; ++v) {
        const int m = mbase + v;
        if (mOK || m < M) {
          float x = acc[t][v] + bv;
          if (act == ACT_LRELU)         x = (x > 0.0f) ? x : 0.01f * x;
          else if (act == ACT_SOFTPLUS) x = (x > 20.0f) ? x : log1pf(expf(x));
          C[(unsigned)(m * N + n)] = x;
        }
      }
    }
  }
}

// ---------------------------------------------------------------------------
// Row LayerNorm (+ leaky relu), in place. One wave32 per row, D <= 256.
// ---------------------------------------------------------------------------
__global__ void ln_lrelu(float* __restrict__ X, const float* __restrict__ g,
                         const float* __restrict__ bt, int rows, int D)
{
  const int wave = threadIdx.x >> 5;
  const int lane = threadIdx.x & 31;
  const int row  = blockIdx.x * (blockDim.x >> 5) + wave;
  if (row >= rows) return;

  float vals[8];
  const int cnt = (D + 31) >> 5;
  float s = 0.0f;
  for (int j = 0; j < cnt; ++j) {
    const int c = lane + (j << 5);
    float v = X[(unsigned)(row * D + imin(c, D - 1))];  // clamped load
    v = (c < D) ? v : 0.0f;
    vals[j] = v;
    s += v;
  }
  const float mean = wave_sum32(s) / (float)D;
  float sq = 0.0f;
  for (int j = 0; j < cnt; ++j) {
    const float d = vals[j] - mean;
    const int c = lane + (j << 5);
    sq += (c < D) ? d * d : 0.0f;
  }
  const float inv = rsqrtf(wave_sum32(sq) / (float)D + 1e-5f);
  for (int j = 0; j < cnt; ++j) {
    const int c = lane + (j << 5);
    if (c < D) {
      float v = (vals[j] - mean) * inv * g[c] + bt[c];
      X[(unsigned)(row * D + c)] = (v > 0.0f) ? v : 0.01f * v;
    }
  }
}

// ---------------------------------------------------------------------------
// S6 selective scan. One wave per (b, d); each lane keeps ceil(300/32)=10
// state elements + A row in registers across the 63 serial steps.
//   h[n] = exp(dt*A[d,n]) * h[n] + (dt*x_t) * B_t[n];  y_t[d] = sum_n C_t[n] h[n]
// Out-of-range lanes (n>=300) use clamped addresses and a 0.0 multiplier.
// ---------------------------------------------------------------------------
__global__ void s6_scan(const float* __restrict__ delta,
                        const float* __restrict__ Bm,
                        const float* __restrict__ Cc,
                        const float* __restrict__ X,
                        const float* __restrict__ Amat,
                        float* __restrict__ Y)
{
  const int b    = blockIdx.x;
  const int wave = threadIdx.x >> 5;
  const int lane = threadIdx.x & 31;
  const int d    = blockIdx.y * (blockDim.x >> 5) + wave;
  if (d >= DM_) return;

  float Ar[10], h[10], msk[10];
  unsigned nc[10];
#pragma unroll
  for (int j = 0; j < 10; ++j) {
    const int n = lane + (j << 5);
    nc[j]  = (unsigned)imin(n, NS_ - 1);
    msk[j] = (n < NS_) ? 1.0f : 0.0f;
    Ar[j]  = Amat[(unsigned)(d * NS_) + nc[j]] * msk[j];
    h[j]   = 0.0f;
  }

  unsigned rowD = (unsigned)(b * L_ * DM_ + d);
  unsigned rowN = (unsigned)(b * L_ * NS_);
  for (int t = 0; t < L_; ++t) {
    const float dt = delta[rowD];
    const float dx = dt * X[rowD];
    float y = 0.0f;
#pragma unroll
    for (int j = 0; j < 10; ++j) {
      const float bm = Bm[rowN + nc[j]] * msk[j];
      const float cv = Cc[rowN + nc[j]];
      h[j] = expf(dt * Ar[j]) * h[j] + dx * bm;
      y += cv * h[j] * msk[j];
    }
    y = wave_sum32(y);
    if (lane == 0) Y[rowD] = y;
    rowD += DM_;
    rowN += NS_;
  }
}

// ---------------------------------------------------------------------------
// Guided cross-modal attention for one batch per block (4 waves):
//   G = softmax(tanh(Q KV^T), -1) @ KV * Q           Q,KV: [63,100]
// KV and the 64x64 score matrix live in LDS; both matmuls use f32 WMMA.
// Clamped addresses; the zeroed score row/col 63 nullifies the padded K.
// ---------------------------------------------------------------------------
__global__ void guided_attn(const float* __restrict__ Q,
                            const float* __restrict__ KV,
                            float* __restrict__ G)
{
  __shared__ float kv_s[L_ * DM_];   // 63*100 floats
  __shared__ float sc[64 * 65];      // padded scores (row stride 65)

  const int b    = blockIdx.x;
  const int tid  = threadIdx.x;
  const int lane = tid & 31;
  const int wave = tid >> 5;
  const unsigned qb = (unsigned)(b * L_ * DM_);
  const float* Qb  = Q  + qb;
  const float* KVb = KV + qb;

  for (int i = tid; i < L_ * DM_; i += blockDim.x) kv_s[i] = KVb[i];
  for (int i = tid; i < 64 * 65;  i += blockDim.x) sc[i] = 0.0f;
  __syncthreads();

  // Phase 1: scores = tanh(Q @ KV^T); wave w owns column tile w (4x4 tiles).
  {
    const int col0 = wave * 16;
    const int koff = (lane >> 4) << 1;
    const int nB   = col0 + (lane & 15);
    const unsigned bB = (unsigned)(imin(nB, L_ - 1) * DM_ + koff);
    for (int ti = 0; ti < 4; ++ti) {
      const int row0 = ti * 16;
      const int m0   = row0 + (lane & 15);
      const unsigned aB = (unsigned)(imin(m0, L_ - 1) * DM_ + koff);
      v8f acc = {};
      for (int k = 0; k < DM_; k += 4) {        // DM_ % 4 == 0: no tail
        v2f a, bf;
        a.x  = Qb[aB + (unsigned)k];
        a.y  = Qb[aB + (unsigned)k + 1u];
        bf.x = kv_s[bB + (unsigned)k];
        bf.y = kv_s[bB + (unsigned)k + 1u];
        acc = __builtin_amdgcn_wmma_f32_16x16x4_f32(false, a, false, bf,
                                                    (short)0, acc, false, false);
      }
      const int mbase = row0 + ((lane >> 4) << 3);
#pragma unroll
      for (int v = 0; v < 8; ++v) {
        const int m = mbase + v;
        if (m < L_ && nB < L_) sc[m * 65 + nB] = tanhf(acc[v]);
      }
    }
  }
  __syncthreads();

  // Phase 2: row softmax over the 63 valid columns (row 63 / col 63 stay 0).
  if (tid < L_) {
    float mx = -1e30f;
    for (int j = 0; j < L_; ++j) mx = fmaxf(mx, sc[tid * 65 + j]);
    float s = 0.0f;
    for (int j = 0; j < L_; ++j) {
      const float e = expf(sc[tid * 65 + j] - mx);
      sc[tid * 65 + j] = e;
      s += e;
    }
    const float inv = 1.0f / s;
    for (int j = 0; j < L_; ++j) sc[tid * 65 + j] *= inv;
  }
  __syncthreads();

  // Phase 3: G = (scores @ KV) * Q ;  4 row-tiles x 7 col-tiles, K padded to 64.
  for (int tile = wave; tile < 4 * 7; tile += 4) {
    const int row0 = (tile / 7) * 16;
    const int col0 = (tile % 7) * 16;
    const int mA   = row0 + (lane & 15);       // <= 63, always valid in sc
    const int nB   = col0 + (lane & 15);
    const int nBc  = imin(nB, DM_ - 1);
    const int koff = (lane >> 4) << 1;
    const int aB   = mA * 65 + koff;
    v8f acc = {};
    for (int k = 0; k < 64; k += 4) {
      const int ka   = k + koff;
      const int kac0 = imin(ka, L_ - 1);
      const int kac1 = imin(ka + 1, L_ - 1);
      v2f a, bf;
      a.x  = sc[aB + k];
      a.y  = sc[aB + k + 1];
      bf.x = kv_s[kac0 * DM_ + nBc];
      bf.y = kv_s[kac1 * DM_ + nBc];
      acc = __builtin_amdgcn_wmma_f32_16x16x4_f32(false, a, false, bf,
                                                  (short)0, acc, false, false);
    }
    const int mbase = row0 + ((lane >> 4) << 3);
#pragma unroll
    for (int v = 0; v < 8; ++v) {
      const int m = mbase + v;
      if (m < L_ && nB < DM_)
        G[qb + (unsigned)(m * DM_ + nB)] = acc[v] * Qb[(unsigned)(m * DM_ + nB)];
    }
  }
}

// ---------------------------------------------------------------------------
// comb[BL,600] = [bi_modal(t,a) | tri_modal(t,a,v) | self(t)] with residuals.
// ---------------------------------------------------------------------------
__global__ void assemble_comb(const float* __restrict__ t,
                              const float* __restrict__ a,
                              const float* __restrict__ v,
                              const float* __restrict__ G0,  // guided(t,a)
                              const float* __restrict__ G1,  // guided(a,t)
                              const float* __restrict__ G2,  // guided(t,v)
                              const float* __restrict__ G3,  // guided(a,v)
                              const float* __restrict__ G4,  // guided(v,t)
                              const float* __restrict__ G5,  // guided(v,a)
                              const float* __restrict__ G6,  // guided(t,t)
                              float* __restrict__ comb)
{
  const int idx = blockIdx.x * blockDim.x + threadIdx.x;
  if (idx >= BL_ * 600) return;
  const int r = idx / 600;
  const int c = idx % 600;
  float val;
  if (c < 100)      { const unsigned k = r * 100 + c;         val = G0[k] + t[k]; }
  else if (c < 200) { const unsigned k = r * 100 + (c - 100); val = G1[k] + a[k]; }
  else if (c < 300) { const unsigned k = r * 100 + (c - 200); val = G0[k] + G2[k] + t[k]; }
  else if (c < 400) { const unsigned k = r * 100 + (c - 300); val = G1[k] + G3[k] + a[k]; }
  else if (c < 500) { const unsigned k = r * 100 + (c - 400); val = G4[k] + G5[k] + v[k]; }
  else              { const unsigned k = r * 100 + (c - 500); val = G6[k] + t[k]; }
  comb[idx] = val;
}

// ---------------------------------------------------------------------------
// Host side
// ---------------------------------------------------------------------------
struct BranchP {
  const float *w_in, *b_in, *g1, *bt1, *w_out, *b_out, *g2, *bt2;
  const float *w1, *b1, *w2, *b2, *w3, *b3, *A;
};

static inline dim3 gemm_grid(int M, int N) {
  return dim3((unsigned)((N + 63) / 64), (unsigned)((M + 63) / 64));
}

static void run_branch(const float* x_in, int din, int dh, const BranchP& p,
                       float* h1, float* h2, float* delta, float* Bm, float* Cc,
                       float* yout, hipStream_t stream)
{
  const dim3 blk(128);
  wmma_gemm_bias_act<<<gemm_grid(BL_, dh), blk, 0, stream>>>(
      x_in, p.w_in, p.b_in, h1, BL_, dh, din, ACT_NONE);
  ln_lrelu<<<(BL_ + 7) / 8, 256, 0, stream>>>(h1, p.g1, p.bt1, BL_, dh);
  wmma_gemm_bias_act<<<gemm_grid(BL_, DM_), blk, 0, stream>>>(
      h1, p.w_out, p.b_out, h2, BL_, DM_, dh, ACT_NONE);
  ln_lrelu<<<(BL_ + 7) / 8, 256, 0, stream>>>(h2, p.g2, p.bt2, BL_, DM_);
  wmma_gemm_bias_act<<<gemm_grid(BL_, DM_), blk, 0, stream>>>(
      h2, p.w1, p.b1, delta, BL_, DM_, DM_, ACT_SOFTPLUS);
  wmma_gemm_bias_act<<<gemm_grid(BL_, NS_), blk, 0, stream>>>(
      h2, p.w2, p.b2, Bm, BL_, NS_, DM_, ACT_NONE);
  wmma_gemm_bias_act<<<gemm_grid(BL_, NS_), blk, 0, stream>>>(
      h2, p.w3, p.b3, Cc, BL_, NS_, DM_, ACT_NONE);
  s6_scan<<<dim3(B_, (DM_ + 7) / 8), 256, 0, stream>>>(delta, Bm, Cc, h2, p.A, yout);
}

extern "C" void kernel_launch(void* const* d_in, const int* in_sizes, int n_in,
                              void* d_out, int out_size, void* d_ws, size_t ws_size,
                              hipStream_t stream)
{
  (void)in_sizes; (void)n_in; (void)out_size; (void)ws_size;
  const float* const* in = (const float* const*)d_in;
  int i = 0;
  const float* text  = in[i++];
  const float* audio = in[i++];
  const float* video = in[i++];

  BranchP pt, pa, pv;
  auto grab = [&](BranchP& p) {
    p.w_in = in[i++]; p.b_in = in[i++]; p.g1 = in[i++]; p.bt1 = in[i++];
    p.w_out = in[i++]; p.b_out = in[i++]; p.g2 = in[i++]; p.bt2 = in[i++];
    p.w1 = in[i++]; p.b1 = in[i++]; p.w2 = in[i++]; p.b2 = in[i++];
    p.w3 = in[i++]; p.b3 = in[i++]; p.A = in[i++];
  };
  grab(pt); grab(pa); grab(pv);
  const float* wo1 = in[i++]; const float* bo1 = in[i++];
  const float* wo2 = in[i++]; const float* bo2 = in[i++];
  const float* wo3 = in[i++]; const float* bo3 = in[i++];
  const float* wo4 = in[i++]; const float* bo4 = in[i++];

  // workspace layout (floats)
  float* ws    = (float*)d_ws;
  float* t_out = ws;                       // BL*100
  float* a_out = t_out + BL_ * DM_;        // BL*100
  float* v_out = a_out + BL_ * DM_;        // BL*100
  float* G     = v_out + BL_ * DM_;        // 7 * BL*100
  float* comb  = G + 7 * BL_ * DM_;        // BL*600
  float* h1    = comb + BL_ * 600;         // BL*256 (max dh)
  float* h2    = h1 + BL_ * 256;           // BL*100
  float* delta = h2 + BL_ * DM_;           // BL*100
  float* Bm    = delta + BL_ * DM_;        // BL*300
  float* Cc    = Bm + BL_ * NS_;           // BL*300
  // G region is dead after assemble_comb -> reuse for the output MLP temps
  float* mlp1  = G;                        // BL*300
  float* mlp2  = G + BL_ * 300;            // BL*300  (fits: G region = BL*700)
  float* mlp3  = h1;                       // BL*50   (h1 dead after branches)

  run_branch(text,  100, 128, pt, h1, h2, delta, Bm, Cc, t_out, stream);
  run_branch(audio,  73, 256, pa, h1, h2, delta, Bm, Cc, a_out, stream);
  run_branch(video, 100, 256, pv, h1, h2, delta, Bm, Cc, v_out, stream);

  float* G0 = G + 0 * BL_ * DM_;  float* G1 = G + 1 * BL_ * DM_;
  float* G2 = G + 2 * BL_ * DM_;  float* G3 = G + 3 * BL_ * DM_;
  float* G4 = G + 4 * BL_ * DM_;  float* G5 = G + 5 * BL_ * DM_;
  float* G6 = G + 6 * BL_ * DM_;

  guided_attn<<<B_, 128, 0, stream>>>(t_out, a_out, G0);
  guided_attn<<<B_, 128, 0, stream>>>(a_out, t_out, G1);
  guided_attn<<<B_, 128, 0, stream>>>(t_out, v_out, G2);
  guided_attn<<<B_, 128, 0, stream>>>(a_out, v_out, G3);
  guided_attn<<<B_, 128, 0, stream>>>(v_out, t_out, G4);
  guided_attn<<<B_, 128, 0, stream>>>(v_out, a_out, G5);
  guided_attn<<<B_, 128, 0, stream>>>(t_out, t_out, G6);

  assemble_comb<<<(BL_ * 600 + 255) / 256, 256, 0, stream>>>(
      t_out, a_out, v_out, G0, G1, G2, G3, G4, G5, G6, comb);

  const dim3 blk(128);
  wmma_gemm_bias_act<<<gemm_grid(BL_, 300), blk, 0, stream>>>(
      comb, wo1, bo1, mlp1, BL_, 300, 600, ACT_LRELU);
  wmma_gemm_bias_act<<<gemm_grid(BL_, 300), blk, 0, stream>>>(
      mlp1, wo2, bo2, mlp2, BL_, 300, 300, ACT_LRELU);
  wmma_gemm_bias_act<<<gemm_grid(BL_, 50), blk, 0, stream>>>(
      mlp2, wo3, bo3, mlp3, BL_, 50, 300, ACT_LRELU);
  wmma_gemm_bias_act<<<gemm_grid(BL_, 2), blk, 0, stream>>>(
      mlp3, wo4, bo4, (float*)d_out, BL_, 2, 50, ACT_NONE);
}